// Gate_1735166788450
// MI455X (gfx1250) — compile-verified
//
#include <hip/hip_runtime.h>
#include <hip/hip_bf16.h>
#include <math.h>

#define DIM       2048
#define N_EXPERTS 64
#define TOPK      6
#define N_TOKENS  32768

#define KC        64                   // K columns per weight chunk
#define WPAD      68                   // padded LDS row stride (floats): 64 + 4 TDM pad
#define WBUF      (N_EXPERTS * WPAD)   // 4352 floats per buffer
#define NCHUNK    (DIM / KC)           // 32
#define SPAD      66                   // score board row stride
#define TOK_PER_WG 128
#define THREADS   256

typedef float v2f  __attribute__((ext_vector_type(2)));
typedef float v4f  __attribute__((ext_vector_type(4)));
typedef float v8f  __attribute__((ext_vector_type(8)));
typedef unsigned int u32x4 __attribute__((ext_vector_type(4)));
typedef int  i32x4 __attribute__((ext_vector_type(4)));
typedef int  i32x8 __attribute__((ext_vector_type(8)));

#if __has_builtin(__builtin_amdgcn_tensor_load_to_lds)
#define USE_TDM 1
#else
#define USE_TDM 0
#endif

#if USE_TDM
// DMA a [64 x KC] fp32 weight tile (row stride DIM) into LDS at lds_byte_addr,
// padding every 64 DWORDs with 4 DWORDs -> LDS row stride WPAD.
__device__ __forceinline__ void tdm_load_weight_chunk(const float* gsrc,
                                                      unsigned lds_byte_addr) {
    const unsigned long long ga = (unsigned long long)(uintptr_t)gsrc;
    u32x4 g0;
    g0.x = 1u;                                            // count=1, user mode
    g0.y = lds_byte_addr;                                 // lds_addr
    g0.z = (unsigned)(ga & 0xFFFFFFFFull);                // global_addr[31:0]
    g0.w = (unsigned)((ga >> 32) & 0x1FFFFFFull)          // global_addr[56:32]
         | (2u << 30);                                    // type = 2 ("image")
    i32x8 g1;
    g1[0] = (int)((2u << 16)        // data_size = 4 bytes
                | (1u << 20)        // pad_enable
                | (5u << 22)        // pad_interval: every 64 DWORDs
                | (3u << 25));      // pad_amount: 4 DWORDs
    g1[1] = (int)(((unsigned)DIM & 0xFFFFu) << 16);       // tensor_dim0[15:0]
    g1[2] = (int)((((unsigned)DIM >> 16) & 0xFFFFu)       // tensor_dim0[31:16]
                | (((unsigned)N_EXPERTS & 0xFFFFu) << 16));// tensor_dim1[15:0]
    g1[3] = (int)(((unsigned)KC) << 16);                  // tile_dim0 = KC
    g1[4] = (int)((unsigned)N_EXPERTS);                   // tile_dim1 = 64, tile_dim2 = 0
    g1[5] = (int)((unsigned)DIM);                         // tensor_dim0_stride[31:0]
    g1[6] = 0;                                            // stride hi / dim1_stride lo
    g1[7] = 0;
    i32x4 z4 = {0, 0, 0, 0};                              // groups 2/3 unused (2-D)
    i32x8 z8 = {0, 0, 0, 0, 0, 0, 0, 0};
    __builtin_amdgcn_tensor_load_to_lds(g0, g1, z4, z4, z8, 0);
}
#endif

__global__ __launch_bounds__(THREADS)
void moe_gate_kernel(const float* __restrict__ x,
                     const float* __restrict__ w,
                     float* __restrict__ out_w,
                     int*   __restrict__ out_i) {
    // Double-buffered weight chunks [2][64][WPAD]; later reused as scores [128][SPAD]
    __shared__ float smem[2 * WBUF];   // 34816 B  (scores need 8448 floats <= 8704)

    const int tid  = threadIdx.x;
    const int lane = tid & 31;
    const int wave = tid >> 5;

    const int tokBase = blockIdx.x * TOK_PER_WG + wave * 16;
    const int m       = lane & 15;           // token/expert index within 16-tile
    const int khalf   = (lane >> 4) << 1;    // lanes 0-15 -> K{0,1}; 16-31 -> K{2,3}

    const float* arow = x + (size_t)(tokBase + m) * DIM + khalf;

    v8f acc0 = {}, acc1 = {}, acc2 = {}, acc3 = {};

#if USE_TDM
    const unsigned ldsBase = (unsigned)(uintptr_t)(&smem[0]);
    if (wave == 0) tdm_load_weight_chunk(w, ldsBase);        // prologue: chunk 0
#endif

    for (int c = 0; c < NCHUNK; ++c) {
        const int kc = c * KC;

#if USE_TDM
        if (wave == 0) __builtin_amdgcn_s_wait_tensorcnt(0); // chunk c arrived
        __syncthreads();                                     // visible to all waves
        if (wave == 0 && (c + 1) < NCHUNK)                   // async-prefetch chunk c+1
            tdm_load_weight_chunk(w + (kc + KC),
                                  ldsBase + (unsigned)(((c + 1) & 1) * (WBUF * 4)));
        const float* wbuf = &smem[(c & 1) * WBUF];
#else
        __syncthreads();
        {   // cooperative staging fallback: 64 rows x KC floats, padded stride WPAD
            const int e    = tid >> 2;      // 0..63
            const int part = tid & 3;       // 16 floats each
            const float* src = w + (size_t)e * DIM + kc + part * 16;
            float*       dst = &smem[e * WPAD + part * 16];
            #pragma unroll
            for (int j = 0; j < 8; ++j)
                *(v2f*)(dst + j * 2) = *(const v2f*)(src + j * 2);
        }
        __syncthreads();
        const float* wbuf = &smem[0];
#endif

        if (kc + KC < DIM)
            __builtin_prefetch(arow + kc + KC, 0, 0);        // next x chunk -> cache

        #pragma unroll 4
        for (int kk = 0; kk < KC; kk += 4) {
            v2f a = *(const v2f*)(arow + kc + kk);

            v2f b0 = *(const v2f*)&wbuf[( 0 + m) * WPAD + kk + khalf];
            acc0 = __builtin_amdgcn_wmma_f32_16x16x4_f32(false, a, false, b0,
                                                         (short)0, acc0, false, false);
            v2f b1 = *(const v2f*)&wbuf[(16 + m) * WPAD + kk + khalf];
            acc1 = __builtin_amdgcn_wmma_f32_16x16x4_f32(false, a, false, b1,
                                                         (short)0, acc1, false, false);
            v2f b2 = *(const v2f*)&wbuf[(32 + m) * WPAD + kk + khalf];
            acc2 = __builtin_amdgcn_wmma_f32_16x16x4_f32(false, a, false, b2,
                                                         (short)0, acc2, false, false);
            v2f b3 = *(const v2f*)&wbuf[(48 + m) * WPAD + kk + khalf];
            acc3 = __builtin_amdgcn_wmma_f32_16x16x4_f32(false, a, false, b3,
                                                         (short)0, acc3, false, false);
        }
    }

    // ---- scatter scores into LDS score board [128][SPAD] (reuses weight LDS) ----
    __syncthreads();
    {
        const int trowBase = wave * 16 + ((lane >> 4) << 3);
        #pragma unroll
        for (int j = 0; j < 8; ++j) {
            const int trow = trowBase + j;
            smem[trow * SPAD +  0 + m] = acc0[j];
            smem[trow * SPAD + 16 + m] = acc1[j];
            smem[trow * SPAD + 32 + m] = acc2[j];
            smem[trow * SPAD + 48 + m] = acc3[j];
        }
    }
    __syncthreads();

    // ---- per-token fp32 softmax + top-6 selection (one thread per token) ----
    if (tid < TOK_PER_WG) {
        float* srow = &smem[tid * SPAD];

        float mx = srow[0];
        #pragma unroll 8
        for (int e = 1; e < N_EXPERTS; ++e) mx = fmaxf(mx, srow[e]);

        float sum = 0.0f;
        #pragma unroll 8
        for (int e = 0; e < N_EXPERTS; ++e) sum += __expf(srow[e] - mx);
        const float inv = 1.0f / sum;          // ROUTE_SCALE == 1.0

        #pragma unroll 8
        for (int e = 0; e < N_EXPERTS; ++e) srow[e] = __expf(srow[e] - mx) * inv;

        const int gtok = blockIdx.x * TOK_PER_WG + tid;
        unsigned long long taken = 0ull;
        #pragma unroll
        for (int s = 0; s < TOPK; ++s) {
            float best = -INFINITY;
            int   bi   = 0;
            for (int e = 0; e < N_EXPERTS; ++e) {
                const bool skip = (taken >> e) & 1ull;
                const float v  = srow[e];
                if (!skip && v > best) { best = v; bi = e; }
            }
            taken |= (1ull << bi);
            out_w[(size_t)gtok * TOPK + s] = best;
            out_i[(size_t)gtok * TOPK + s] = bi;
        }
    }
}

extern "C" void kernel_launch(void* const* d_in, const int* in_sizes, int n_in,
                              void* d_out, int out_size, void* d_ws, size_t ws_size,
                              hipStream_t stream) {
    (void)in_sizes; (void)n_in; (void)out_size; (void)d_ws; (void)ws_size;
    const float* x = (const float*)d_in[0];   // [N_TOKENS, DIM]
    const float* w = (const float*)d_in[1];   // [N_EXPERTS, DIM]

    float* out_w = (float*)d_out;                              // [N_TOKENS, TOPK] fp32
    int*   out_i = (int*)(out_w + (size_t)N_TOKENS * TOPK);    // [N_TOKENS, TOPK] i32

    dim3 grid(N_TOKENS / TOK_PER_WG);
    moe_gate_kernel<<<grid, THREADS, 0, stream>>>(x, w, out_w, out_i);
}